// Yolov2_20693152432222
// MI455X (gfx1250) — compile-verified
//
#include <hip/hip_runtime.h>
#include <hip/hip_bf16.h>

#define NUM_CLASSES 80
#define MTOT        300000
#define TOPK        1000
#define CONF_THRESH 0.001f
#define NMS_THRESH  0.5f
#define CLS_OFF     4096.0f

// ---------------- workspace layout (4-byte units) ----------------
#define WS_SCORES   0                      // 300000 f32
#define WS_LABELS   300000                 // 300000 u32
#define WS_HIST     600000                 // 65536 u32
#define HIST_SIZE   65536
#define WS_SUP      (WS_HIST + HIST_SIZE)  // 1000*32 u32 suppression bit-matrix
#define WS_CNT      (WS_SUP + 32000)       // [0]=cand count, [1]=threshold bucket
#define CAND_CAP    4096
#define WS_CANDBITS (WS_CNT + 16)
#define WS_CANDIDX  (WS_CANDBITS + CAND_CAP)
#define WS_TKSCORE  (WS_CANDIDX + CAND_CAP)   // 1000 f32
#define WS_TKIDX    (WS_TKSCORE + TOPK)       // 1000 u32
#define WS_BOXK     (WS_TKIDX + TOPK)         // 1000*4 f32 (raw boxes, 16B aligned)
#define WS_LABK     (WS_BOXK + 4*TOPK)        // 1000 u32
#define WS_AREAK    (WS_LABK + TOPK)          // 1000 f32
#define ZERO_COUNT  (HIST_SIZE + 32000 + 16)  // hist+sup+counters are contiguous

// ---------------- gfx1250 feature guards ----------------
#if defined(__AMDGCN__) && __has_builtin(__builtin_amdgcn_global_load_async_to_lds_b128)
#define HAVE_ASYNC_LDS 1
typedef int async_v4i __attribute__((vector_size(4 * sizeof(int))));
typedef __attribute__((address_space(1))) async_v4i gv4i_t;   // prints as __device__ v4i
typedef __attribute__((address_space(3))) async_v4i lv4i_t;   // LDS v4i
#else
#define HAVE_ASYNC_LDS 0
#endif

#if defined(__AMDGCN__) && __has_builtin(__builtin_amdgcn_wmma_f32_16x16x4_f32)
#define HAVE_WMMA4 1
#else
#define HAVE_WMMA4 0
#endif

typedef float v2f __attribute__((ext_vector_type(2)));
typedef float v8f __attribute__((ext_vector_type(8)));

__device__ __forceinline__ void wait_async_lds() {
#if defined(__AMDGCN__)
#if __has_builtin(__builtin_amdgcn_s_wait_asynccnt)
  __builtin_amdgcn_s_wait_asynccnt(0);
#else
  asm volatile("s_wait_asynccnt 0" ::: "memory");
#endif
#endif
}

// ---------------- kernel 0: zero hist + suppression matrix + counters ----------------
__global__ void k0_zero(unsigned* wu) {
  int g = blockIdx.x * blockDim.x + threadIdx.x;
  if (g < ZERO_COUNT) wu[WS_HIST + g] = 0u;
}

// ---------------- kernel 1: fused score + argmax (streaming, async->LDS) ----------------
// block=128 threads, 128 rows per block. LDS tile [128][84] (rows 16B aligned, padded).
__global__ void k1_score(const float* __restrict__ obj, const float* __restrict__ cls,
                         float* wf, unsigned* wu) {
  __shared__ float tile[128][84];
  const int tid = threadIdx.x;
  const int R0  = blockIdx.x * 128;

  // stage 128x80 floats: 20 iterations of 128 lanes x float4, perfectly coalesced
#pragma unroll 4
  for (int it = 0; it < 20; ++it) {
    int idx = it * 128 + tid;
    int f = idx * 4;
    int r = f / 80, c = f % 80;     // c is a multiple of 4 -> 16B aligned in LDS
    int g = R0 + r;
    if (g < MTOT) {
      const float* src = cls + (size_t)g * NUM_CLASSES + c;
#if HAVE_ASYNC_LDS
      __builtin_amdgcn_global_load_async_to_lds_b128(
          (gv4i_t*)src, (lv4i_t*)&tile[r][c], 0, 0);
#else
      *(float4*)&tile[r][c] = *(const float4*)src;
#endif
    }
  }
#if HAVE_ASYNC_LDS
  wait_async_lds();
#endif
  __syncthreads();

  int g = R0 + tid;
  if (g < MTOT) {
    // argmax on raw logits (sigmoid/sqrt are monotone)
    float best = tile[tid][0];
    int bi = 0;
#pragma unroll 8
    for (int k = 1; k < NUM_CLASSES; ++k) {
      float v = tile[tid][k];
      if (v > best) { best = v; bi = k; }
    }
    float so = 1.0f / (1.0f + __expf(-obj[g]));
    float sc = 1.0f / (1.0f + __expf(-best));
    float s  = sqrtf(so * sc);
    wf[WS_SCORES + g] = s;
    wu[WS_LABELS + g] = (unsigned)bi;
    unsigned b = __float_as_uint(s) >> 14;   // s in (0,1): bucket < 65536
    atomicAdd(&wu[WS_HIST + b], 1u);
  }
}

// ---------------- kernel 2: find threshold bucket ----------------
__global__ void k2_thresh(unsigned* wu) {
  __shared__ unsigned csum[1024];
  int tid = threadIdx.x;
  unsigned s = 0;
  for (int b = 0; b < 64; ++b) s += wu[WS_HIST + tid * 64 + b];
  csum[tid] = s;
  __syncthreads();
  if (tid == 0) {
    unsigned run = 0, T = 0;
    for (int c = 1023; c >= 0; --c) {
      unsigned cs = csum[c];
      if (run + cs >= (unsigned)TOPK) {
        unsigned acc = run;
        for (int b = c * 64 + 63; b >= c * 64; --b) {
          acc += wu[WS_HIST + b];
          if (acc >= (unsigned)TOPK) { T = (unsigned)b; break; }
        }
        break;
      }
      run += cs;
    }
    wu[WS_CNT + 1] = T;
  }
}

// ---------------- kernel 3: compact candidates >= threshold bucket ----------------
__global__ void k3_compact(const float* wf, unsigned* wu) {
  int g = blockIdx.x * blockDim.x + threadIdx.x;
  if (g >= MTOT) return;
  unsigned T = wu[WS_CNT + 1];
  unsigned bits = __float_as_uint(wf[WS_SCORES + g]);
  if ((bits >> 14) >= T) {
    unsigned pos = atomicAdd(&wu[WS_CNT + 0], 1u);
    if (pos < CAND_CAP) {
      wu[WS_CANDBITS + pos] = bits;
      wu[WS_CANDIDX + pos]  = (unsigned)g;
    }
  }
}

// ---------------- kernel 4: bitonic top-1000 (desc score, asc index) + gather ----------------
__global__ void k4_sort(const float* __restrict__ box, float* wf, unsigned* wu) {
  __shared__ unsigned long long skey[CAND_CAP];   // 32 KB
  int tid = threadIdx.x;
  unsigned cnt = wu[WS_CNT + 0];
  if (cnt > CAND_CAP) cnt = CAND_CAP;
  for (int i = tid; i < CAND_CAP; i += 1024) {
    unsigned long long key = 0ull;
    if ((unsigned)i < cnt) {
      unsigned bits = wu[WS_CANDBITS + i];
      unsigned idx  = wu[WS_CANDIDX + i];        // idx < 2^19
      key = ((unsigned long long)bits << 20) | (unsigned long long)(0xFFFFFu - idx);
    }
    skey[i] = key;
  }
  __syncthreads();
  for (int k = 2; k <= CAND_CAP; k <<= 1) {
    for (int j = k >> 1; j > 0; j >>= 1) {
      for (int i = tid; i < CAND_CAP; i += 1024) {
        int ixj = i ^ j;
        if (ixj > i) {
          unsigned long long a = skey[i], b = skey[ixj];
          bool desc = ((i & k) == 0);
          if (desc ? (a < b) : (a > b)) { skey[i] = b; skey[ixj] = a; }
        }
      }
      __syncthreads();
    }
  }
  if (tid < TOPK) {
    unsigned long long key = skey[tid];
    float s; unsigned idx;
    if (key != 0ull) {
      s   = __uint_as_float((unsigned)(key >> 20));
      idx = 0xFFFFFu - (unsigned)(key & 0xFFFFFu);
    } else { s = 0.0f; idx = 0u; }
    wf[WS_TKSCORE + tid] = s;
    wu[WS_TKIDX + tid]   = idx;
    float4 b = *(const float4*)(box + (size_t)idx * 4);
    *(float4*)&wf[WS_BOXK + 4 * tid] = b;
    wu[WS_LABK + tid] = wu[WS_LABELS + idx];
    float w = fmaxf(b.z - b.x, 0.0f), h = fmaxf(b.w - b.y, 0.0f);
    wf[WS_AREAK + tid] = w * h;
  }
}

// ---------------- kernel 5: suppression bit-matrix, one wave per 16x16 tile ----------------
// area_i + area_j computed as a rank-2 WMMA: A=[area_i,1] (16x4), B=[1;area_j] (4x16)
__global__ void k5_sup(const float* wf, unsigned* wu) {
  int gw   = (blockIdx.x * blockDim.x + threadIdx.x) >> 5;   // uniform per wave
  int lane = threadIdx.x & 31;
  if (gw >= 63 * 63) return;
  int ti = gw / 63, tj = gw % 63;
  int i0 = ti * 16, j0 = tj * 16;
  int n = lane & 15, hi = lane >> 4;

  float aI = (i0 + n < TOPK) ? wf[WS_AREAK + i0 + n] : 0.0f;
  float aJ = (j0 + n < TOPK) ? wf[WS_AREAK + j0 + n] : 0.0f;

  v8f asum;
#if HAVE_WMMA4
  {
    v2f A, B;
    A.x = hi ? 0.0f : aI;   A.y = hi ? 0.0f : 1.0f;   // K0=area_i, K1=1 (lanes 0-15)
    B.x = hi ? aJ   : 1.0f; B.y = 0.0f;               // K0 row = ones, K1 row = area_j
    v8f c = {};
    asum = __builtin_amdgcn_wmma_f32_16x16x4_f32(false, A, false, B, (short)0, c,
                                                 false, false);
  }
#else
#pragma unroll
  for (int v = 0; v < 8; ++v) {
    int i = i0 + v + 8 * hi;
    float ai = (i < TOPK) ? wf[WS_AREAK + i] : 0.0f;
    asum[v] = ai + aJ;
  }
#endif

  int j = j0 + n;
  bool jv = j < TOPK;
  float4 bj = make_float4(0.f, 0.f, 0.f, 0.f);
  float offJ = 0.0f;
  if (jv) {
    bj = *(const float4*)&wf[WS_BOXK + 4 * j];
    offJ = (float)wu[WS_LABK + j] * CLS_OFF;
  }
  float x1j = bj.x + offJ, y1j = bj.y + offJ, x2j = bj.z + offJ, y2j = bj.w + offJ;

#pragma unroll
  for (int v = 0; v < 8; ++v) {
    int i = i0 + v + 8 * hi;
    bool pred = false;
    if ((i < TOPK) && jv) {
      float4 bi = *(const float4*)&wf[WS_BOXK + 4 * i];
      float offI = (float)wu[WS_LABK + i] * CLS_OFF;
      float xx1 = fmaxf(bi.x + offI, x1j);
      float yy1 = fmaxf(bi.y + offI, y1j);
      float xx2 = fminf(bi.z + offI, x2j);
      float yy2 = fminf(bi.w + offI, y2j);
      float inter = fmaxf(xx2 - xx1, 0.0f) * fmaxf(yy2 - yy1, 0.0f);
      float iou = inter / (asum[v] - inter + 1e-9f);
      pred = iou > NMS_THRESH;
    }
    unsigned long long m = __ballot(pred ? 1 : 0);   // wave32: low 32 bits
    unsigned lo   = (unsigned)(m & 0xFFFFull);          // rows M=v     (lanes 0-15)
    unsigned hi16 = (unsigned)((m >> 16) & 0xFFFFull);  // rows M=v+8   (lanes 16-31)
    if (lane == 0) {
      int w = j0 >> 5, sh = j0 & 31;   // sh in {0,16}
      int iA = i0 + v, iB = i0 + v + 8;
      if (iA < TOPK && lo)   atomicOr(&wu[WS_SUP + iA * 32 + w], lo << sh);
      if (iB < TOPK && hi16) atomicOr(&wu[WS_SUP + iB * 32 + w], hi16 << sh);
    }
  }
}

// ---------------- kernel 6: greedy scan (single wave, bitmask) + outputs ----------------
__global__ void k6_scan_out(const float* wf, const unsigned* wu, float* out) {
  __shared__ unsigned keepW[32];
  int tid = threadIdx.x;
  if (tid < 32) {
    unsigned kw = 0;
    for (int b = 0; b < 32; ++b) {
      int j = tid * 32 + b;
      if (j < TOPK && wf[WS_TKSCORE + j] > CONF_THRESH) kw |= (1u << b);
    }
    for (int i = 0; i < TOPK; ++i) {
      int w = i >> 5, bit = i & 31;
      unsigned wv = __shfl(kw, w, 32);
      if ((wv >> bit) & 1u) {
        unsigned sup = wu[WS_SUP + i * 32 + tid];
        unsigned m;
        if (tid < w)       m = 0u;
        else if (tid == w) m = (bit == 31) ? 0u : (0xFFFFFFFFu << (bit + 1));
        else               m = 0xFFFFFFFFu;
        kw &= ~(sup & m);
      }
    }
    keepW[tid] = kw;
  }
  __syncthreads();
  for (int t = tid; t < TOPK; t += blockDim.x) {
    unsigned k = (keepW[t >> 5] >> (t & 31)) & 1u;
    float kf = (float)k;
    float s = wf[WS_TKSCORE + t];
    float4 b = *(const float4*)&wf[WS_BOXK + 4 * t];
    out[t * 5 + 0] = s * kf;
    out[t * 5 + 1] = b.x * kf;
    out[t * 5 + 2] = b.y * kf;
    out[t * 5 + 3] = b.z * kf;
    out[t * 5 + 4] = b.w * kf;
    out[5 * TOPK + t] = (float)wu[WS_LABK + t];   // labels_k
    out[6 * TOPK + t] = kf;                       // keep
  }
}

extern "C" void kernel_launch(void* const* d_in, const int* in_sizes, int n_in,
                              void* d_out, int out_size, void* d_ws, size_t ws_size,
                              hipStream_t stream) {
  (void)in_sizes; (void)n_in; (void)out_size; (void)ws_size;
  const float* obj = (const float*)d_in[0];
  const float* cls = (const float*)d_in[1];
  const float* box = (const float*)d_in[2];
  float*    out = (float*)d_out;
  float*    wf  = (float*)d_ws;
  unsigned* wu  = (unsigned*)d_ws;

  k0_zero   <<<(ZERO_COUNT + 255) / 256, 256, 0, stream>>>(wu);
  k1_score  <<<(MTOT + 127) / 128, 128, 0, stream>>>(obj, cls, wf, wu);
  k2_thresh <<<1, 1024, 0, stream>>>(wu);
  k3_compact<<<(MTOT + 255) / 256, 256, 0, stream>>>(wf, wu);
  k4_sort   <<<1, 1024, 0, stream>>>(box, wf, wu);
  k5_sup    <<<(63 * 63 + 7) / 8, 256, 0, stream>>>(wf, wu);
  k6_scan_out<<<1, 1024, 0, stream>>>(wf, wu, out);
}